// SpectralConvWavelet3D_16853451669623
// MI455X (gfx1250) — compile-verified
//
#include <hip/hip_runtime.h>

// ---------------- db4 filters (pywt convention, pre-flipped for correlation) --
#define H0 0.2303778133088552f
#define H1 0.7148465705525415f
#define H2 0.6308807679295904f
#define H3 (-0.02798376941698385f)
#define H4 (-0.18703481171888114f)
#define H5 0.030841381835986965f
#define H6 0.032883011666982945f
#define H7 (-0.010597401784997278f)

// forward: lo[j] = sum_t FLO[t]*x[(2j+t-6) mod n],  hi likewise with FHI
__constant__ float FLO[8] = { H0,  H1,  H2,  H3,  H4,  H5,  H6,  H7};
__constant__ float FHI[8] = {-H0,  H1, -H2,  H3, -H4,  H5, -H6,  H7};
// inverse (conv-transpose stride 2, pad 7, trim 6):
// y[k] = sum_{t: (k+t)&1==1, j=(k+t-1)/2 < nc} ILO[t]*lo[j] + IHI[t]*hi[j]
__constant__ float ILO[8] = { H7,  H6,  H5,  H4,  H3,  H2,  H1,  H0};
__constant__ float IHI[8] = { H7, -H6,  H5, -H4,  H3, -H2,  H1, -H0};

typedef float v2f __attribute__((ext_vector_type(2)));
typedef float v8f __attribute__((ext_vector_type(8)));

// ---------------- problem constants ------------------------------------------
#define BATCH 4
#define CH    64
#define BC    256          // BATCH*CH
#define NP    9261         // 21*21*21
#define CS    2370816      // BC*NP, one subband
#define A35   3951360      // BC*21*21*35
#define D35   6585600      // BC*21*35*35

// =============================================================================
// Forward DWT kernels (periodic padding), optional hi-band output
// =============================================================================
__global__ void fwd_dwt_d(const float* __restrict__ in, float* __restrict__ lo,
                          float* __restrict__ hi, int n, int nc, int sh, int sw,
                          long total, int writeHi) {
  long t = (long)blockIdx.x * blockDim.x + threadIdx.x;
  if (t >= total) return;
  int w = (int)(t % sw); long r = t / sw;
  int h = (int)(r % sh); r /= sh;
  int j = (int)(r % nc); int bc = (int)(r / nc);
  float aLo = 0.f, aHi = 0.f;
#pragma unroll
  for (int tap = 0; tap < 8; ++tap) {
    int d = (2 * j + tap - 6 + n) % n;
    float v = in[(((long)bc * n + d) * sh + h) * sw + w];
    aLo += FLO[tap] * v;
    aHi += FHI[tap] * v;
  }
  long o = (((long)bc * nc + j) * sh + h) * sw + w;
  lo[o] = aLo;
  if (writeHi) hi[o] = aHi;
}

__global__ void fwd_dwt_h(const float* __restrict__ in, float* __restrict__ lo,
                          float* __restrict__ hi, int n, int nc, int sd, int sw,
                          long total, int writeHi) {
  long t = (long)blockIdx.x * blockDim.x + threadIdx.x;
  if (t >= total) return;
  int w = (int)(t % sw); long r = t / sw;
  int j = (int)(r % nc); r /= nc;
  int d = (int)(r % sd); int bc = (int)(r / sd);
  float aLo = 0.f, aHi = 0.f;
#pragma unroll
  for (int tap = 0; tap < 8; ++tap) {
    int hh = (2 * j + tap - 6 + n) % n;
    float v = in[(((long)bc * sd + d) * n + hh) * sw + w];
    aLo += FLO[tap] * v;
    aHi += FHI[tap] * v;
  }
  long o = (((long)bc * sd + d) * nc + j) * sw + w;
  lo[o] = aLo;
  if (writeHi) hi[o] = aHi;
}

__global__ void fwd_dwt_w(const float* __restrict__ in, float* __restrict__ lo,
                          float* __restrict__ hi, int n, int nc, int sd, int sh,
                          long total, int writeHi) {
  long t = (long)blockIdx.x * blockDim.x + threadIdx.x;
  if (t >= total) return;
  int j = (int)(t % nc); long r = t / nc;
  int h = (int)(r % sh); r /= sh;
  int d = (int)(r % sd); int bc = (int)(r / sd);
  float aLo = 0.f, aHi = 0.f;
  long ibase = (((long)bc * sd + d) * sh + h) * n;
#pragma unroll
  for (int tap = 0; tap < 8; ++tap) {
    int ww = (2 * j + tap - 6 + n) % n;
    float v = in[ibase + ww];
    aLo += FLO[tap] * v;
    aHi += FHI[tap] * v;
  }
  long o = (((long)bc * sd + d) * sh + h) * nc + j;
  lo[o] = aLo;
  if (writeHi) hi[o] = aHi;
}

// =============================================================================
// Inverse DWT kernels (conv-transpose stride2), hi may be nullptr (zero details)
// =============================================================================
__global__ void inv_dwt_w(const float* __restrict__ lo, const float* __restrict__ hi,
                          float* __restrict__ out, int nc, int olen, int sd, int sh,
                          long total) {
  long t = (long)blockIdx.x * blockDim.x + threadIdx.x;
  if (t >= total) return;
  int k = (int)(t % olen); long r = t / olen;
  int h = (int)(r % sh); r /= sh;
  int d = (int)(r % sd); int bc = (int)(r / sd);
  long ibase = (((long)bc * sd + d) * sh + h) * nc;
  float y = 0.f;
#pragma unroll
  for (int tap = 0; tap < 8; ++tap) {
    if (((k + tap) & 1) == 1) {
      int j = (k + tap - 1) >> 1;
      if (j < nc) {
        y += ILO[tap] * lo[ibase + j];
        if (hi) y += IHI[tap] * hi[ibase + j];
      }
    }
  }
  out[(((long)bc * sd + d) * sh + h) * olen + k] = y;
}

__global__ void inv_dwt_h(const float* __restrict__ lo, const float* __restrict__ hi,
                          float* __restrict__ out, int nc, int olen, int sd, int sw,
                          long total) {
  long t = (long)blockIdx.x * blockDim.x + threadIdx.x;
  if (t >= total) return;
  int w = (int)(t % sw); long r = t / sw;
  int k = (int)(r % olen); r /= olen;
  int d = (int)(r % sd); int bc = (int)(r / sd);
  long ibase = (((long)bc * sd + d) * nc) * sw + w;
  float y = 0.f;
#pragma unroll
  for (int tap = 0; tap < 8; ++tap) {
    if (((k + tap) & 1) == 1) {
      int j = (k + tap - 1) >> 1;
      if (j < nc) {
        y += ILO[tap] * lo[ibase + (long)j * sw];
        if (hi) y += IHI[tap] * hi[ibase + (long)j * sw];
      }
    }
  }
  out[(((long)bc * sd + d) * olen + k) * sw + w] = y;
}

__global__ void inv_dwt_d(const float* __restrict__ lo, const float* __restrict__ hi,
                          float* __restrict__ out, int nc, int olen, int sh, int sw,
                          long total) {
  long t = (long)blockIdx.x * blockDim.x + threadIdx.x;
  if (t >= total) return;
  int w = (int)(t % sw); long r = t / sw;
  int h = (int)(r % sh); r /= sh;
  int k = (int)(r % olen); int bc = (int)(r / olen);
  float y = 0.f;
#pragma unroll
  for (int tap = 0; tap < 8; ++tap) {
    if (((k + tap) & 1) == 1) {
      int j = (k + tap - 1) >> 1;
      if (j < nc) {
        long idx = (((long)bc * nc + j) * sh + h) * sw + w;
        y += ILO[tap] * lo[idx];
        if (hi) y += IHI[tap] * hi[idx];
      }
    }
  }
  out[(((long)bc * olen + k) * sh + h) * sw + w] = y;
}

// =============================================================================
// Channel mixing:  out[s][b][o][p] = sum_i coeff[s][b][i][p] * w[s][i][o][p]
// Per voxel p this is Out(64x4) = W_p^T(64x64) x C_p(64x4), done with
// V_WMMA_F32_16X16X4_F32:  M = Cout (4 tiles of 16), K = Cin (16 steps of 4),
// N = 16 lanes with B=4 valid columns.  Weight (p-innermost in memory) is
// staged through 128 KB of LDS per workgroup (8 voxels x 16 KB), each of the
// 8 waves then runs a full 64-WMMA GEMM for its voxel from LDS.
// =============================================================================
__global__ __launch_bounds__(256)
void mix_wmma(const float* __restrict__ coeff, const float* __restrict__ wt,
              float* __restrict__ outc) {
  __shared__ float wlds[8 * 4096];          // 128 KB: wlds[pj][i*64 + o]
  const int s  = blockIdx.y;
  const int p0 = blockIdx.x * 8;
  const float* wsub = wt + (size_t)s * 64 * 64 * NP;

  // cooperative stage of 8 voxels' (i,o) weight slabs, p innermost for
  // contiguous 32B global segments; adjacent blocks share lines via L2
#pragma unroll 4
  for (int it = 0; it < 128; ++it) {
    int flat = threadIdx.x + (it << 8);     // 0 .. 32767
    int pj   = flat & 7;
    int io   = flat >> 3;                   // i*64 + o
    int p    = p0 + pj;
    wlds[(pj << 12) + io] = (p < NP) ? wsub[(size_t)io * NP + p] : 0.f;
  }
  __syncthreads();

  const int wave = threadIdx.x >> 5;        // 8 waves, one voxel each
  const int lane = threadIdx.x & 31;
  const int p    = p0 + wave;
  if (p >= NP) return;                      // wave-uniform, EXEC stays full

  const int nn    = lane & 15;              // N index (batch b) / M row half
  const int khalf = lane >> 4;              // 0: K=0,1   1: K=2,3
  const float* cs = coeff + (size_t)s * (BATCH * CH * NP);

  // B operand in registers: B[k][n] = c[b=n][i=k][p] (n<4 valid, rest zero)
  v2f breg[16];
#pragma unroll
  for (int k0 = 0; k0 < 16; ++k0) {
    float b0 = 0.f, b1 = 0.f;
    if (nn < 4) {
      const size_t base = ((size_t)(nn * CH + (k0 * 4 + 2 * khalf))) * NP + p;
      b0 = cs[base];
      b1 = cs[base + NP];
    }
    breg[k0].x = b0;
    breg[k0].y = b1;
  }

  float* os = outc + (size_t)s * (BATCH * CH * NP);
  const float* wp = wlds + (wave << 12);
#pragma unroll
  for (int mt = 0; mt < 4; ++mt) {          // Cout tiles of 16
    v8f acc = {0.f, 0.f, 0.f, 0.f, 0.f, 0.f, 0.f, 0.f};
#pragma unroll
    for (int k0 = 0; k0 < 16; ++k0) {       // Cin in steps of 4
      const int ib = k0 * 4 + 2 * khalf;    // A lane layout: K pair per half
      v2f a;
      a.x = wp[ib * 64 + mt * 16 + nn];     // A[m][k]   = w[i=k][o=mt*16+m]
      a.y = wp[(ib + 1) * 64 + mt * 16 + nn];
      acc = __builtin_amdgcn_wmma_f32_16x16x4_f32(
          false, a, false, breg[k0], (short)0, acc, false, false);
    }
    if (nn < 4) {                           // D: N=lane&15, M=v+8*khalf
#pragma unroll
      for (int v = 0; v < 8; ++v) {
        const int m = v + 8 * khalf;
        os[((size_t)(nn * CH + mt * 16 + m)) * NP + p] = acc[v];
      }
    }
  }
}

// =============================================================================
// Host orchestration: ping-pong through two 36,700,160-float scratch regions
// =============================================================================
extern "C" void kernel_launch(void* const* d_in, const int* in_sizes, int n_in,
                              void* d_out, int out_size, void* d_ws, size_t ws_size,
                              hipStream_t stream) {
  (void)in_sizes; (void)n_in; (void)out_size; (void)ws_size;
  const float* x  = (const float*)d_in[0];   // (4,64,64,64,64)
  const float* wt = (const float*)d_in[1];   // (8,64,64,21,21,21)
  float* out = (float*)d_out;                // (4,64,64,64,64)

  const long REG = 36700160;                 // BC*35*64*64 floats
  float* reg0 = (float*)d_ws;
  float* reg1 = reg0 + REG;

  const int TPB = 256;
  auto nb = [&](long n) { return (unsigned)((n + TPB - 1) / TPB); };

  // ---- forward low-pass chain to level-1 cA (only path that survives) ----
  long t1 = (long)BC * 35 * 64 * 64;   // 36,700,160
  fwd_dwt_d<<<nb(t1), TPB, 0, stream>>>(x, reg0, nullptr, 64, 35, 64, 64, t1, 0);
  long t2 = (long)BC * 35 * 35 * 64;   // 20,070,400
  fwd_dwt_h<<<nb(t2), TPB, 0, stream>>>(reg0, reg1, nullptr, 64, 35, 35, 64, t2, 0);
  long t3 = (long)BC * 35 * 35 * 35;   // 10,976,000
  fwd_dwt_w<<<nb(t3), TPB, 0, stream>>>(reg1, reg0, nullptr, 64, 35, 35, 35, t3, 0);

  // ---- level-2 full 8-subband decomposition (35 -> 21 per axis) ----
  long t4 = (long)BC * 21 * 35 * 35;   // 6,585,600
  fwd_dwt_d<<<nb(t4), TPB, 0, stream>>>(reg0, reg1, reg1 + D35, 35, 21, 35, 35, t4, 1);
  long t5 = (long)BC * 21 * 21 * 35;   // 3,951,360
  for (int f = 0; f < 2; ++f)
    fwd_dwt_h<<<nb(t5), TPB, 0, stream>>>(reg1 + (long)f * D35,
                                          reg0 + (long)(2 * f) * A35,
                                          reg0 + (long)(2 * f + 1) * A35,
                                          35, 21, 21, 35, t5, 1);
  long t6 = (long)BC * NP;             // 2,370,816
  for (int q = 0; q < 4; ++q)
    fwd_dwt_w<<<nb(t6), TPB, 0, stream>>>(reg0 + (long)q * A35,
                                          reg1 + (long)(2 * q) * CS,
                                          reg1 + (long)(2 * q + 1) * CS,
                                          35, 21, 21, 21, t6, 1);

  // ---- per-voxel channel mixing on the WMMA pipe (coeff reg1 -> outc reg0) --
  dim3 mg((NP + 7) / 8, 8);            // 1158 x 8 subbands
  mix_wmma<<<mg, 256, 0, stream>>>(reg1, wt, reg0);

  // ---- level-2 reconstruction (21 -> 35 per axis) ----
  for (int q = 0; q < 4; ++q)
    inv_dwt_w<<<nb(t5), TPB, 0, stream>>>(reg0 + (long)(2 * q) * CS,
                                          reg0 + (long)(2 * q + 1) * CS,
                                          reg1 + (long)q * A35,
                                          21, 35, 21, 21, t5);
  for (int r = 0; r < 2; ++r)
    inv_dwt_h<<<nb(t4), TPB, 0, stream>>>(reg1 + (long)(2 * r) * A35,
                                          reg1 + (long)(2 * r + 1) * A35,
                                          reg0 + (long)r * D35,
                                          21, 35, 21, 35, t4);
  inv_dwt_d<<<nb(t3), TPB, 0, stream>>>(reg0, reg0 + D35, reg1, 21, 35, 35, 35, t3);

  // ---- level-1 reconstruction with zero details (lo-only, 35 -> 64) ----
  inv_dwt_w<<<nb(t2), TPB, 0, stream>>>(reg1, nullptr, reg0, 35, 64, 35, 35, t2);
  inv_dwt_h<<<nb(t1), TPB, 0, stream>>>(reg0, nullptr, reg1, 35, 64, 35, 64, t1);
  long to = (long)BC * 64 * 64 * 64;   // 67,108,864
  inv_dwt_d<<<nb(to), TPB, 0, stream>>>(reg1, nullptr, out, 35, 64, 64, 64, to);
}